// ContinuousLearningLayer_42167988912993
// MI455X (gfx1250) — compile-verified
//
#include <hip/hip_runtime.h>
#include <hip/hip_bf16.h>
#include <stdint.h>

// Problem constants (reference: in (2,512) -> N=1024 flat, w (512,512) -> M=262144 flat)
#define N_ROWS   1024
#define M_COLS   262144
#define RADIUS   25                   // window 51, pad 25
#define ROWS_PER_BLOCK 4
#define WAVES_PER_BLOCK 8
#define WORDS_PER_WAVE 32             // 1024 columns per wave
#define SUPER_COLS (WAVES_PER_BLOCK * WORDS_PER_WAVE * 32)   // 8192
#define CHUNK_ELEMS (SUPER_COLS + 96)                         // cols [c0-64, c0+8224) = 8288 floats
#define THRESH   0.1f
#define SENTINEL 3.0e30f

typedef unsigned int uint32;
typedef uint32 v4u __attribute__((ext_vector_type(4)));
typedef int    v4i __attribute__((ext_vector_type(4)));
typedef int    v8i __attribute__((ext_vector_type(8)));

// ---------------------------------------------------------------------------
// TDM: async DMA of a 1-D run of `len` fp32 elements  global -> LDS.
// D# per CDNA5 ISA ch.8: count=1, type=2, data_size=4B, tensor_dim0=tile_dim0=len,
// tensor_dim1=1, no padding/iterate/gather. Tracked by TENSORcnt.
// ---------------------------------------------------------------------------
__device__ __forceinline__ void tdm_load_1d_f32(const float* gsrc, uint32 ldsByteOff, uint32 len)
{
    uint64_t ga = (uint64_t)(uintptr_t)gsrc;
    v4u g0;
    g0[0] = 1u;                                   // [1:0]  count=1, user mode
    g0[1] = ldsByteOff;                           // [63:32] lds_addr (bytes)
    g0[2] = (uint32)(ga & 0xFFFFFFFFu);           // [95:64] global_addr lo
    g0[3] = (uint32)((ga >> 32) & 0x1FFFFFFu)     // [120:96] global_addr hi
          | (2u << 30);                           // [127:126] type = 2 ("image")
    v8i g1{};
    g1[0] = (int)(2u << 16);                      // data_size=2 -> 4 bytes; wg_mask=0
    g1[1] = (int)((len & 0xFFFFu) << 16);         // tensor_dim0[15:0]  (bits 63:48)
    g1[2] = (int)(((len >> 16) & 0xFFFFu)         // tensor_dim0[31:16] (bits 79:64)
          |  (1u << 16));                         // tensor_dim1 = 1    (bits 95:80)
    g1[3] = (int)((len & 0xFFFFu) << 16);         // tile_dim0 = len    (bits 127:112)
    g1[4] = 0;                                    // tile_dim1=0 (unused), tile_dim2=0
    g1[5] = (int)len;                             // tensor_dim0_stride lo
    g1[6] = 0;                                    // stride hi, dim1_stride lo
    g1[7] = 0;
    v4i z{};
#if __clang_major__ >= 23
    __builtin_amdgcn_tensor_load_to_lds(g0, g1, z, z, v8i{}, 0);
#else
    __builtin_amdgcn_tensor_load_to_lds(g0, g1, z, z, 0);
#endif
}

// v_writelane_b32: write a scalar value into one lane of a VGPR.
// Value and lane-select are both scalar operands per the ISA (SSRC0/SSRC1).
__device__ __forceinline__ int writelane(uint32 sval, int k, int vdst)
{
    asm("v_writelane_b32 %0, %1, %2" : "+v"(vdst) : "s"(sval), "s"(k));
    return vdst;
}

// Causal width-51 OR-dilation: C bit i of word K = OR mask[32K+i-50 .. 32K+i],
// from mask words (w0,w1,w2) = (K-2, K-1, K).  Used scalar (for C32) and
// per-lane vectorized (32 words at once on VALU).
__device__ __forceinline__ uint32 cword(uint32 w0, uint32 w1, uint32 w2)
{
    uint64_t v = ((uint64_t)w2 << 32) | (uint64_t)w1;
    v |= v << 1;  v |= v << 2;  v |= v << 4;      // widths 2,4,8
    v |= v << 8;  v |= v << 16;                   // widths 16,32
    v |= v << 19;                                 // width 51
    uint32 top = (uint32)(v >> 32);
    uint32 s = w0;                                // halo from word K-2: suffix-OR
    s |= s >> 1;  s |= s >> 2;  s |= s >> 4;  s |= s >> 8;  s |= s >> 16;
    return top | (s >> 14);                       // contribution[i] = OR S[i+14..31]
}

// Expand one uniform 32-bit mask word into 32 lanes of {0.0f, 1.0f} with a
// single VALU op: v_cndmask_b32 selects per lane on bit[lane] of the scalar.
__device__ __forceinline__ float expand_bit(uint32 ow)
{
    float val;
    asm("v_cndmask_b32 %0, 0, 1.0, %1" : "=v"(val) : "s"(ow));
    return val;
}

__global__ __launch_bounds__(WAVES_PER_BLOCK * 32)
void pooled_threshold_kernel(const float* __restrict__ in,
                             const float* __restrict__ w,
                             float* __restrict__ out)
{
    __shared__ float ldsW[CHUNK_ELEMS];           // 33152 B of the 320 KB WGP LDS

    const int lane  = (int)(threadIdx.x & 31);
    const int wave  = (int)(threadIdx.x >> 5);    // 0..7
    const int super = (int)blockIdx.x;            // 0..31 column supertiles
    const int n0    = (int)blockIdx.y * ROWS_PER_BLOCK;
    const int c0    = super * SUPER_COLS;         // first output column of supertile

    // Chunk covers logical columns [c0-64, c0+8224); TDM loads the clamp to [0, M).
    int gstart = c0 - 64;              if (gstart < 0)      gstart = 0;
    int gend   = c0 + SUPER_COLS + 32; if (gend > M_COLS)   gend = M_COLS;
    const int F = gstart - (c0 - 64);             // front halo slots not covered by TDM
    const int T = gend   - (c0 - 64);             // first tail halo slot not covered

    // Sentinel-fill the uncovered halo (edge blocks only; disjoint from TDM target).
    for (int i = (int)threadIdx.x; i < F; i += WAVES_PER_BLOCK * 32)
        ldsW[i] = SENTINEL;
    for (int i = T + (int)threadIdx.x; i < CHUNK_ELEMS; i += WAVES_PER_BLOCK * 32)
        ldsW[i] = SENTINEL;

    // Async-stage the weight chunk via the Tensor Data Mover (one wave issues).
    if (wave == 0) {
        uint32 ldsOff = (uint32)F * 4u + (uint32)(uintptr_t)(&ldsW[0]);
        tdm_load_1d_f32(w + gstart, ldsOff, (uint32)(gend - gstart));
        __builtin_amdgcn_s_wait_tensorcnt(0);
    }
    __syncthreads();

    // ---- per-wave tile: 32 output words starting at global word j0g
    const int j0g = super * (WAVES_PER_BLOCK * WORDS_PER_WAVE) + wave * WORDS_PER_WAVE;

    // Preload this wave's 35 weight values per lane (mask words j0g-2 .. j0g+32).
    float wvv[35];
#pragma unroll
    for (int t = 0; t < 35; ++t) {
        int col = (j0g - 2 + t) * 32 + lane;      // may be <0 or >=M: maps to sentinel
        wvv[t] = ldsW[col - (c0 - 64)];           // index always within [0, CHUNK_ELEMS)
    }

    // bpermute byte-addresses for +1 / +2 lane shifts (wrap lanes are patched).
    const int a1 = 4 * ((lane + 1) & 31);
    const int a2 = 4 * ((lane + 2) & 31);

    for (int r = 0; r < ROWS_PER_BLOCK; ++r) {
        const float x = in[n0 + r];

        // 32 threshold bits per v_cmp: wave32 ballot packs one word per compare.
        uint32 mw[35];
#pragma unroll
        for (int t = 0; t < 35; ++t)
            mw[t] = __builtin_amdgcn_ballot_w32(__builtin_fabsf(wvv[t] - x) < THRESH);

        // Distribute mask words across lanes: lane k holds mw[k] (k = 0..31).
        int vA = 0;
#pragma unroll
        for (int k = 0; k < 32; ++k)
            vA = writelane(mw[k], k, vA);

        // Neighbors: lane k gets mw[k+1] / mw[k+2]; wrap lanes patched from SGPRs.
        int vB = __builtin_amdgcn_ds_bpermute(a1, vA);
        vB = writelane(mw[32], 31, vB);
        int vC = __builtin_amdgcn_ds_bpermute(a2, vA);
        vC = writelane(mw[32], 30, vC);
        vC = writelane(mw[33], 31, vC);

        // Vectorized dilation: lane k computes C word j0g+k (one pass on VALU
        // covers all 32 words; ~1 instruction per word instead of ~24).
        uint32 Ck = cword((uint32)vA, (uint32)vB, (uint32)vC);

        // C_{k+1} into lane k; lane 31 patched with scalar C32 (words 30,31,32).
        uint32 C32 = cword(mw[32], mw[33], mw[34]);
        int vCn = __builtin_amdgcn_ds_bpermute(a1, (int)Ck);
        vCn = writelane(C32, 31, vCn);

        // out bit b = C[32j + b + 25]  ->  ow_k = (C_k >> 25) | (C_{k+1} << 7)
        uint32 ow = (Ck >> RADIUS) | ((uint32)vCn << (32 - RADIUS));

        float* __restrict__ p = out + (size_t)(n0 + r) * (size_t)M_COLS
                                    + (size_t)j0g * 32 + lane;

        // Pull each word back to an SGPR and expand with one v_cndmask per word.
#pragma unroll
        for (int k = 0; k < 32; ++k) {
            uint32 sw = (uint32)__builtin_amdgcn_readlane((int)ow, k);
            __builtin_nontemporal_store(expand_bit(sw), p + k * 32);
        }
    }
}

extern "C" void kernel_launch(void* const* d_in, const int* in_sizes, int n_in,
                              void* d_out, int out_size, void* d_ws, size_t ws_size,
                              hipStream_t stream)
{
    (void)in_sizes; (void)n_in; (void)out_size; (void)d_ws; (void)ws_size;
    const float* in = (const float*)d_in[0];   // input_features, 1024 fp32 flat
    const float* w  = (const float*)d_in[1];   // weight_matrix, 262144 fp32 flat
    float* out      = (float*)d_out;           // 1024 x 262144 fp32

    dim3 grid(M_COLS / SUPER_COLS, N_ROWS / ROWS_PER_BLOCK);   // 32 x 256
    dim3 block(WAVES_PER_BLOCK * 32);                          // 256 = 8 wave32
    pooled_threshold_kernel<<<grid, block, 0, stream>>>(in, w, out);
}